// MixMeansCovsLayer_87084756894001
// MI455X (gfx1250) — compile-verified
//
#include <hip/hip_runtime.h>
#include <math.h>

typedef __attribute__((ext_vector_type(2))) float v2f;
typedef __attribute__((ext_vector_type(8))) float v8f;

#define N_STATES 25
#define N_CH     80
#define BATCH    64
#define TIME     400
#define M_TOTAL  (BATCH * TIME)          // 25600 rows
#define N_TOTAL  (N_CH + N_CH * N_CH)    // 6480 cols = [mu | D]
#define KPAD     28                      // K=25 padded to multiple of 4
#define KK_STEPS (KPAD / 4)              // 7 wmma k-steps
#define NTILES   (N_TOTAL / 16)          // 405
#define NT_PER_WAVE 5                    // 405 = 81 * 5
#define NGROUPS  (NTILES / NT_PER_WAVE)  // 81
#define MTILES   (M_TOTAL / 16)          // 1600
#define BSW_ELEMS (NTILES * KK_STEPS * 64)  // 181,440 floats (~726 KB)

// ---------------------------------------------------------------------------
// Prep: build softplus-scaled, zero-padded B = [mu | D] (KPAD x N_TOTAL) in
// d_ws, pre-swizzled into the exact V_WMMA_F32_16X16X4_F32 B-fragment lane
// layout:  Bsw[((tile*7 + kk)*32 + lane)*2 + e]
//   lane 0..15  : N = tile*16 + lane,    K = kk*4 + {0,1}
//   lane 16..31 : N = tile*16 + lane-16, K = kk*4 + {2,3}
// so a wave's B fragment is one contiguous 256-byte v2f load.
// ---------------------------------------------------------------------------
__global__ void mixmc_prep_B(const float* __restrict__ mu,
                             const float* __restrict__ Dm,
                             const float* __restrict__ ascale,
                             float* __restrict__ Bsw) {
    int idx = blockIdx.x * blockDim.x + threadIdx.x;
    if (idx >= BSW_ELEMS) return;
    int e    = idx & 1;
    int lane = (idx >> 1) & 31;
    int rest = idx >> 6;
    int kk   = rest % KK_STEPS;
    int t    = rest / KK_STEPS;

    int k   = kk * 4 + ((lane >> 4) << 1) + e;
    int col = t * 16 + (lane & 15);

    float val = 0.0f;
    if (k < N_STATES) {
        float x = ascale[k];
        float s = (x > 20.0f) ? x : log1pf(expf(x));   // softplus, stable
        float src = (col < N_CH) ? mu[k * N_CH + col]
                                 : Dm[(size_t)k * (N_CH * N_CH) + (col - N_CH)];
        val = s * src;
    }
    Bsw[idx] = val;
}

// Store one wave's 5 accumulated 16x16 tiles with a compile-time row stride,
// through a uniform base pointer + 32-bit per-lane offset (saddr-form stores,
// r*LD folded into the instruction immediate).
template <int LD>
__device__ __forceinline__ void
store_tiles(float* __restrict__ base, int m0, int g, int colShift, int lane,
            const v8f* acc) {
    const int colLocal = lane & 15;
    const int rowOff   = (lane >> 4) * 8;
    const unsigned rowBase = (unsigned)(m0 + rowOff) * (unsigned)LD;
#pragma unroll
    for (int nt = 0; nt < NT_PER_WAVE; ++nt) {
        unsigned col = (unsigned)((g * NT_PER_WAVE + nt) * 16 + colLocal - colShift);
        unsigned off = rowBase + col;
#pragma unroll
        for (int r = 0; r < 8; ++r)
            base[off + (unsigned)(r * LD)] = acc[nt][r];
    }
}

// ---------------------------------------------------------------------------
// GEMM: [25600 x 28] x [28 x 6480] via V_WMMA_F32_16X16X4_F32.
// One wave = one 16-row M tile x five 16-col N tiles (35 WMMAs).
// Group 0 writes m_t (ld=80); groups 1..80 write C_t (ld=6400).
// ---------------------------------------------------------------------------
__global__ __launch_bounds__(256) void
mixmc_gemm(const float* __restrict__ alpha,
           const float* __restrict__ Bsw,
           float* __restrict__ out_m,
           float* __restrict__ out_C) {
    const int lane = threadIdx.x & 31;
    // Wave work id is uniform per wave: force it scalar so tile indices,
    // output base and the m_t/C_t branch live in SGPRs.
    const int w = __builtin_amdgcn_readfirstlane(
        (int)(blockIdx.x * 8 + (threadIdx.x >> 5)));
    const int m = w % MTILES;   // consecutive waves: consecutive M tiles,
    const int g = w / MTILES;   // same N group -> B frags shared in-block

    const int m0    = m * 16;
    const int row   = m0 + (lane & 15);
    const int kbase = (lane >> 4) * 2;

    // Load the wave's A fragments once (16 rows x KPAD, zero-padded K>=25).
    const float* __restrict__ arow = alpha + (size_t)row * N_STATES;
    v2f afrag[KK_STEPS];
#pragma unroll
    for (int kk = 0; kk < KK_STEPS; ++kk) {
        int k0 = kk * 4 + kbase;
        float x = (k0     < N_STATES) ? arow[k0]     : 0.0f;
        float y = (k0 + 1 < N_STATES) ? arow[k0 + 1] : 0.0f;
        afrag[kk].x = x;
        afrag[kk].y = y;
    }

    const v2f* __restrict__ bfr = (const v2f*)Bsw;

    v8f acc[NT_PER_WAVE];
#pragma unroll
    for (int nt = 0; nt < NT_PER_WAVE; ++nt) {
        const int t = g * NT_PER_WAVE + nt;
        v8f c = {};
#pragma unroll
        for (int kk = 0; kk < KK_STEPS; ++kk) {
            v2f b = bfr[(size_t)(t * KK_STEPS + kk) * 32 + lane];
            // D(16x16,f32) = A(16x4,f32) * B(4x16,f32) + C
            c = __builtin_amdgcn_wmma_f32_16x16x4_f32(
                    false, afrag[kk], false, b, (short)0, c, false, false);
        }
        acc[nt] = c;
    }

    // C/D layout — VGPR r, lanes 0-15: row m0+r, col n0+lane;
    //             lanes 16-31: row m0+8+r, col n0+lane-16.
    // g is scalar, and tiles never straddle the m_t/C_t boundary (col 80).
    if (g == 0) {
        store_tiles<N_CH>(out_m, m0, g, 0, lane, acc);
    } else {
        store_tiles<N_CH * N_CH>(out_C, m0, g, N_CH, lane, acc);
    }
}

extern "C" void kernel_launch(void* const* d_in, const int* in_sizes, int n_in,
                              void* d_out, int out_size, void* d_ws, size_t ws_size,
                              hipStream_t stream) {
    const float* alpha  = (const float*)d_in[0];  // [64,400,25]
    const float* mu     = (const float*)d_in[1];  // [25,80]
    const float* Dm     = (const float*)d_in[2];  // [25,80,80]
    const float* ascale = (const float*)d_in[3];  // [25]

    float* out_m = (float*)d_out;                           // [25600,80]
    float* out_C = out_m + (size_t)M_TOTAL * N_CH;          // [25600,6400]
    float* Bsw   = (float*)d_ws;                            // 726 KB swizzled B

    mixmc_prep_B<<<(BSW_ELEMS + 255) / 256, 256, 0, stream>>>(mu, Dm, ascale, Bsw);

    const int total_waves = MTILES * NGROUPS;               // 129,600
    mixmc_gemm<<<total_waves / 8, 256, 0, stream>>>(alpha, Bsw, out_m, out_C);
}